// GCNModel_47476568490264
// MI455X (gfx1250) — compile-verified
//
#include <hip/hip_runtime.h>
#include <math.h>

// ---------------------------------------------------------------------------
// GCN (3x GCNConv + fused MLP head) for MI455X / gfx1250, wave32.
// Dense matmuls use V_WMMA_F32_16X16X4_F32 (full fp32 accuracy; matmul FLOPs
// are tiny vs the edge-scatter atomic phase, so no reason to drop precision).
// FC1(32->1024)+ReLU+FC2(1024->1)+sigmoid fused per 16-node tile in registers.
// Scatter uses native global_atomic_add_f32 via unsafeAtomicAdd (avoids any
// CAS-loop lowering of atomicAdd(float*)).
// ---------------------------------------------------------------------------

typedef float v2f __attribute__((ext_vector_type(2)));
typedef float v8f __attribute__((ext_vector_type(8)));

#define N_NODES 50000
#define N_EDGES 800000
#define NODE_TILES (N_NODES / 16)   // 3125, exact

__device__ __forceinline__ float relu_nc(float x) { return x > 0.0f ? x : 0.0f; }

// ---------------- utility kernels ----------------

__global__ void zero_f32(float* __restrict__ p, int n) {
    int i = blockIdx.x * blockDim.x + threadIdx.x;
    if (i < n) p[i] = 0.0f;
}

__global__ void count_deg(const int* __restrict__ dst, float* __restrict__ deg) {
    int e = blockIdx.x * blockDim.x + threadIdx.x;
    if (e < N_EDGES) unsafeAtomicAdd(&deg[dst[e]], 1.0f);
}

__global__ void deg_finish(const float* __restrict__ deg,
                           float* __restrict__ dinv, float* __restrict__ inv) {
    int i = blockIdx.x * blockDim.x + threadIdx.x;
    if (i < N_NODES) {
        float d = deg[i] + 1.0f;     // +1 self loop
        dinv[i] = rsqrtf(d);
        inv[i]  = 1.0f / d;
    }
}

// ---------------- dense linear: H = X @ W  (no bias), via f32 WMMA ----------
// One wave computes a 16(nodes) x 16(cols) tile, accumulating K in steps of 4.
// A frag (16x4 f32): lane<16 -> (K+0,K+1) of row M=lane; lane>=16 -> (K+2,K+3).
// B frag (4x16 f32): VGPR r = row K+r (lanes0-15) / K+r+2 (lanes16-31), N=lane%16.
// C/D (16x16 f32):   VGPR r = row M=r (lanes0-15) / M=r+8 (lanes16-31), N=lane%16.

template <int CIN, int COUT>
__global__ void linear_wmma(const float* __restrict__ X,
                            const float* __restrict__ W,
                            float* __restrict__ H) {
    const int tilesC = COUT / 16;
    int wave = (blockIdx.x * blockDim.x + threadIdx.x) >> 5;
    int lane = threadIdx.x & 31;
    if (wave >= NODE_TILES * tilesC) return;            // wave-uniform exit
    int tn   = wave / tilesC;
    int tc   = wave % tilesC;
    int l15  = lane & 15;
    int half = lane >> 4;
    int row  = tn * 16 + l15;
    int colb = tc * 16;

    v8f c = {0.f, 0.f, 0.f, 0.f, 0.f, 0.f, 0.f, 0.f};
#pragma unroll
    for (int k = 0; k < CIN; k += 4) {
        v2f a = *(const v2f*)(X + (size_t)row * CIN + k + 2 * half);
        v2f b;
        b.x = W[(size_t)(k + 2 * half + 0) * COUT + colb + l15];
        b.y = W[(size_t)(k + 2 * half + 1) * COUT + colb + l15];
        c = __builtin_amdgcn_wmma_f32_16x16x4_f32(
                /*neg_a=*/false, a, /*neg_b=*/false, b,
                /*c_mod=*/(short)0, c, /*reuse_a=*/false, /*reuse_b=*/false);
    }
#pragma unroll
    for (int r = 0; r < 8; ++r) {
        int m = r + 8 * half;
        H[(size_t)(tn * 16 + m) * COUT + colb + l15] = c[r];
    }
}

// ---------------- edge scatter: agg[dst] += h[src] * dinv[src]*dinv[dst] ----
// One thread per (edge, 4-channel quad): float4 gather + 4 native f32 atomics.
// agg arrays (<=12.8MB) are L2-resident (192MB L2), so atomics resolve at L2.

template <int COUT>
__global__ void scatter_edges(const int* __restrict__ src,
                              const int* __restrict__ dst,
                              const float* __restrict__ H,
                              const float* __restrict__ dinv,
                              float* __restrict__ AGG) {
    const int NQ = COUT / 4;
    long i = (long)blockIdx.x * blockDim.x + threadIdx.x;
    if (i >= (long)N_EDGES * NQ) return;
    int e = (int)(i / NQ);
    int q = (int)(i % NQ);
    int s = src[e], d = dst[e];
    float nrm = dinv[s] * dinv[d];
    float4 v = *(const float4*)(H + (size_t)s * COUT + 4 * q);
    float* ag = AGG + (size_t)d * COUT + 4 * q;
    unsafeAtomicAdd(ag + 0, v.x * nrm);
    unsafeAtomicAdd(ag + 1, v.y * nrm);
    unsafeAtomicAdd(ag + 2, v.z * nrm);
    unsafeAtomicAdd(ag + 3, v.w * nrm);
}

// ---------------- finish: out = relu(agg + h * (1/deg) + b) -----------------

template <int COUT>
__global__ void gcn_finish(const float* __restrict__ AGG,
                           const float* __restrict__ H,
                           const float* __restrict__ inv,
                           const float* __restrict__ b,
                           float* __restrict__ OUT) {
    int i = blockIdx.x * blockDim.x + threadIdx.x;
    if (i >= N_NODES * COUT) return;
    int n = i / COUT, cch = i % COUT;
    float v = AGG[i] + H[i] * inv[n] + b[cch];
    OUT[i] = relu_nc(v);
}

// ---------------- fused head: sigmoid(relu(h@Wf1+bf1)@Wf2+bf2) --------------
// One wave per 16 nodes. A frags (h, 16x32) live in registers; loop over 64
// column tiles of Wf1; each 16x16 FC1 tile is ReLU'd and dotted with Wf2 in
// registers -- the 205MB FC1 activation never touches memory.

__global__ void fc_fused(const float* __restrict__ H,    // [N,32]
                         const float* __restrict__ Wf1,  // [32,1024]
                         const float* __restrict__ bf1,  // [1024]
                         const float* __restrict__ Wf2,  // [1024]
                         const float* __restrict__ bf2,  // [1]
                         float* __restrict__ OUT) {      // [N]
    int wave = (blockIdx.x * blockDim.x + threadIdx.x) >> 5;
    int lane = threadIdx.x & 31;
    if (wave >= NODE_TILES) return;                      // wave-uniform exit
    int l15  = lane & 15;
    int half = lane >> 4;
    int row  = wave * 16 + l15;

    v2f a[8];
#pragma unroll
    for (int k = 0; k < 8; ++k)
        a[k] = *(const v2f*)(H + (size_t)row * 32 + 4 * k + 2 * half);

    float acc[8] = {0.f, 0.f, 0.f, 0.f, 0.f, 0.f, 0.f, 0.f};

    for (int t = 0; t < 64; ++t) {
        int colb = t * 16;
        float bias = bf1[colb + l15];
        v8f c = {bias, bias, bias, bias, bias, bias, bias, bias};
#pragma unroll
        for (int k = 0; k < 8; ++k) {
            v2f b;
            b.x = Wf1[(size_t)(4 * k + 2 * half + 0) * 1024 + colb + l15];
            b.y = Wf1[(size_t)(4 * k + 2 * half + 1) * 1024 + colb + l15];
            c = __builtin_amdgcn_wmma_f32_16x16x4_f32(
                    false, a[k], false, b, (short)0, c, false, false);
        }
        float w2 = Wf2[colb + l15];
#pragma unroll
        for (int r = 0; r < 8; ++r)
            acc[r] += relu_nc(c[r]) * w2;
    }

    // reduce over N (16 lanes per half); xor masks 1,2,4,8 stay in each half
#pragma unroll
    for (int r = 0; r < 8; ++r) {
        float s = acc[r];
        s += __shfl_xor(s, 1, 32);
        s += __shfl_xor(s, 2, 32);
        s += __shfl_xor(s, 4, 32);
        s += __shfl_xor(s, 8, 32);
        acc[r] = s;
    }
    if (l15 == 0) {
        float b2 = bf2[0];
#pragma unroll
        for (int r = 0; r < 8; ++r) {
            int node = wave * 16 + 8 * half + r;
            float z = acc[r] + b2;
            OUT[node] = 1.0f / (1.0f + expf(-z));
        }
    }
}

// ---------------------------------------------------------------------------

static inline int cdiv(long a, long b) { return (int)((a + b - 1) / b); }

extern "C" void kernel_launch(void* const* d_in, const int* in_sizes, int n_in,
                              void* d_out, int out_size, void* d_ws, size_t ws_size,
                              hipStream_t stream) {
    const float* x   = (const float*)d_in[0];
    const int*   ei  = (const int*)d_in[1];
    const float* W1  = (const float*)d_in[2];
    const float* b1  = (const float*)d_in[3];
    const float* W2  = (const float*)d_in[4];
    const float* b2  = (const float*)d_in[5];
    const float* W3  = (const float*)d_in[6];
    const float* b3  = (const float*)d_in[7];
    const float* Wf1 = (const float*)d_in[8];
    const float* bf1 = (const float*)d_in[9];
    const float* Wf2 = (const float*)d_in[10];
    const float* bf2 = (const float*)d_in[11];
    float* out = (float*)d_out;

    const int* src = ei;             // edge_index[0]
    const int* dst = ei + N_EDGES;   // edge_index[1]

    // workspace carve (floats), offsets aligned to 64 floats (256B)
    float* ws   = (float*)d_ws;
    float* dinv = ws;                         // 50000
    float* inv  = ws + 50048;                 // 50000
    float* deg  = ws + 100096;                // 50000
    float* h    = ws + 150144;                // up to 50000*64
    float* agg  = h   + 3200000;
    float* bufA = agg + 3200000;
    float* bufB = bufA + 3200000;

    const int TB = 256;

    // degrees
    zero_f32<<<cdiv(N_NODES, TB), TB, 0, stream>>>(deg, N_NODES);
    count_deg<<<cdiv(N_EDGES, TB), TB, 0, stream>>>(dst, deg);
    deg_finish<<<cdiv(N_NODES, TB), TB, 0, stream>>>(deg, dinv, inv);

    // ---- layer 1: 16 -> 32 ----
    {
        long waves = (long)NODE_TILES * (32 / 16);
        linear_wmma<16, 32><<<cdiv(waves * 32, TB), TB, 0, stream>>>(x, W1, h);
        zero_f32<<<cdiv((long)N_NODES * 32, TB), TB, 0, stream>>>(agg, N_NODES * 32);
        scatter_edges<32><<<cdiv((long)N_EDGES * 8, TB), TB, 0, stream>>>(src, dst, h, dinv, agg);
        gcn_finish<32><<<cdiv((long)N_NODES * 32, TB), TB, 0, stream>>>(agg, h, inv, b1, bufA);
    }
    // ---- layer 2: 32 -> 64 ----
    {
        long waves = (long)NODE_TILES * (64 / 16);
        linear_wmma<32, 64><<<cdiv(waves * 32, TB), TB, 0, stream>>>(bufA, W2, h);
        zero_f32<<<cdiv((long)N_NODES * 64, TB), TB, 0, stream>>>(agg, N_NODES * 64);
        scatter_edges<64><<<cdiv((long)N_EDGES * 16, TB), TB, 0, stream>>>(src, dst, h, dinv, agg);
        gcn_finish<64><<<cdiv((long)N_NODES * 64, TB), TB, 0, stream>>>(agg, h, inv, b2, bufB);
    }
    // ---- layer 3: 64 -> 32 ----
    {
        long waves = (long)NODE_TILES * (32 / 16);
        linear_wmma<64, 32><<<cdiv(waves * 32, TB), TB, 0, stream>>>(bufB, W3, h);
        zero_f32<<<cdiv((long)N_NODES * 32, TB), TB, 0, stream>>>(agg, N_NODES * 32);
        scatter_edges<32><<<cdiv((long)N_EDGES * 8, TB), TB, 0, stream>>>(src, dst, h, dinv, agg);
        gcn_finish<32><<<cdiv((long)N_NODES * 32, TB), TB, 0, stream>>>(agg, h, inv, b3, bufA);
    }
    // ---- fused MLP head ----
    {
        long waves = NODE_TILES;
        fc_fused<<<cdiv(waves * 32, TB), TB, 0, stream>>>(bufA, Wf1, bf1, Wf2, bf2, out);
    }
}